// TimeHGraphConvLayer_78159814853227
// MI455X (gfx1250) — compile-verified
//
#include <hip/hip_runtime.h>

// ---------------- problem constants (match reference) ----------------
#define SVCN   50000
#define INSTN  100000
#define NODEN  50000
#define FDIM   128
#define HDIM   128
#define OUTD   256
#define TOTAL  (SVCN + NODEN + INSTN)          // 200000 concat rows
#define E_SC   400000
#define E_IN   100000
#define E_NI   100000
#define E_II   800000

// ---------------- workspace layout (float offsets) ----------------
#define OFF_CAT   0LL                                   // TOTAL*128 = 25,600,000
#define OFF_HSC   (OFF_CAT  + (long long)TOTAL * 128)
#define OFF_HIN   (OFF_HSC  + (long long)SVCN  * 128)
#define OFF_HNI   (OFF_HIN  + (long long)INSTN * 128)
#define OFF_HII   (OFF_HNI  + (long long)NODEN * 128)
#define OFF_DEG   (OFF_HII  + (long long)INSTN * 128)
#define D_NS_SC   (OFF_DEG + 0LL)         // SVCN
#define D_ND_SC   (D_NS_SC + SVCN)        // SVCN
#define D_NS_IN   (D_ND_SC + SVCN)        // INSTN
#define D_ND_IN   (D_NS_IN + INSTN)       // NODEN
#define D_NS_NI   (D_ND_IN + NODEN)       // NODEN
#define D_ND_NI   (D_NS_NI + NODEN)       // INSTN
#define D_NS_II   (D_ND_NI + INSTN)       // INSTN
#define D_ND_II   (D_NS_II + INSTN)       // INSTN
#define DEG_TOTAL 600000

typedef __attribute__((ext_vector_type(2))) float v2f;
typedef __attribute__((ext_vector_type(8))) float v8f;

__device__ __forceinline__ float lrelu(float x) { return x > 0.f ? x : 0.01f * x; }

// Hardware f32 atomic add (global_atomic_add_f32), not a CAS loop: the edge
// scatter issues ~179M of these and is the bandwidth-dominant phase.
__device__ __forceinline__ void atomAddF(float* p, float v) { unsafeAtomicAdd(p, v); }

// ---------------- init kernels ----------------
__global__ void init_deg_kernel(float* deg) {
    int i = blockIdx.x * blockDim.x + threadIdx.x;
    if (i < DEG_TOTAL) deg[i] = 0.f;
}

// cat[row][:] initialized with the bias of its region (b_ni + b_ii for the
// inst region); destination-side normalization is folded into the edge
// scatter, so after the scatter pass cat holds the full pre-activation.
__global__ void init_cat_kernel(float* __restrict__ cat,
                                const float* __restrict__ b_sc,
                                const float* __restrict__ b_in,
                                const float* __restrict__ b_ni,
                                const float* __restrict__ b_ii) {
    long long tid = (long long)blockIdx.x * blockDim.x + threadIdx.x;
    if (tid >= (long long)TOTAL * 32) return;
    int row = (int)(tid >> 5);
    int c4  = (int)(tid & 31) * 4;
    float4 v;
    if (row < SVCN) {
        v = *(const float4*)&b_sc[c4];
    } else if (row < SVCN + NODEN) {
        v = *(const float4*)&b_in[c4];
    } else {
        float4 a = *(const float4*)&b_ni[c4];
        float4 b = *(const float4*)&b_ii[c4];
        v = make_float4(a.x + b.x, a.y + b.y, a.z + b.z, a.w + b.w);
    }
    *(float4*)&cat[(long long)row * 128 + c4] = v;
}

__global__ void init_out_kernel(float* __restrict__ out, const float* __restrict__ b_tot) {
    int i = blockIdx.x * blockDim.x + threadIdx.x;
    if (i < OUTD * HDIM) out[i] = b_tot[i >> 7];   // out[o][j] = b_tot[o]
}

// ---------------- degree accumulation + rsqrt ----------------
__global__ void degree_kernel(const int* __restrict__ src, const int* __restrict__ dst,
                              float* __restrict__ degS, float* __restrict__ degD, int nE) {
    int e = blockIdx.x * blockDim.x + threadIdx.x;
    if (e >= nE) return;
    atomAddF(&degS[src[e]], 1.f);
    atomAddF(&degD[dst[e]], 1.f);
}

__global__ void rsqrt_kernel(float* deg) {
    int i = blockIdx.x * blockDim.x + threadIdx.x;
    if (i < DEG_TOTAL) {
        float d = deg[i];
        deg[i] = rsqrtf(d > 1.f ? d : 1.f);
    }
}

// ---------------- projection: H = (X @ W) * ns[:,None]  (fp32 WMMA) ----------------
// Block: 256 threads = 8 waves. Block covers 64 rows of X, each wave owns one
// 16-column tile of the 128-wide output and iterates 4 row tiles.
// W (128x128, 64KB) is staged in LDS once per block.
__global__ __launch_bounds__(256) void proj_kernel(const float* __restrict__ X,
                                                   const float* __restrict__ W,
                                                   const float* __restrict__ ns,
                                                   float* __restrict__ Hout, int nrows) {
    __shared__ float Wlds[128 * 128];
    int t = threadIdx.x;
    for (int i = t * 4; i < 128 * 128; i += 256 * 4)
        *(float4*)&Wlds[i] = *(const float4*)&W[i];
    __syncthreads();

    const int lane = t & 31;
    const int wave = t >> 5;        // col tile 0..7
    const int half = lane >> 4;     // 0/1
    const int l16  = lane & 15;
    const int r0   = blockIdx.x * 64;
    const int n0   = wave * 16;

    // K-invariant A-row base pointers (clamped; OOB rows masked at store)
    const float* Arow[4];
#pragma unroll
    for (int rt = 0; rt < 4; ++rt) {
        int row = r0 + rt * 16 + l16;
        int rc  = row < nrows ? row : nrows - 1;
        Arow[rt] = X + (long long)rc * 128;
    }

    v8f acc[4] = {v8f{}, v8f{}, v8f{}, v8f{}};

    for (int k = 0; k < 128; k += 4) {
        const int kk = k + 2 * half;
        v2f b;
        b.x = Wlds[kk * 128 + n0 + l16];
        b.y = Wlds[(kk + 1) * 128 + n0 + l16];
#pragma unroll
        for (int rt = 0; rt < 4; ++rt) {
            v2f a = *(const v2f*)&Arow[rt][kk];
            acc[rt] = __builtin_amdgcn_wmma_f32_16x16x4_f32(
                false, a, false, b, (short)0, acc[rt], false, false);
        }
    }

    if (r0 + 64 <= nrows) {
        // fast path: block fully in range, no per-row guards
#pragma unroll
        for (int rt = 0; rt < 4; ++rt) {
#pragma unroll
            for (int i = 0; i < 8; ++i) {
                int row = r0 + rt * 16 + i + 8 * half;
                Hout[(long long)row * 128 + n0 + l16] = acc[rt][i] * ns[row];
            }
        }
    } else {
#pragma unroll
        for (int rt = 0; rt < 4; ++rt) {
#pragma unroll
            for (int i = 0; i < 8; ++i) {
                int row = r0 + rt * 16 + i + 8 * half;
                if (row < nrows)
                    Hout[(long long)row * 128 + n0 + l16] = acc[rt][i] * ns[row];
            }
        }
    }
}

// ---------------- edge scatter: cat[dst] += H[src] * nd[dst] ----------------
// One wave (32 lanes) per edge, float4 per lane. nd (dest normalization) is
// folded in here so no separate normalization pass is needed.
__global__ void scatter_kernel(const float* __restrict__ H,
                               const int* __restrict__ src, const int* __restrict__ dst,
                               const float* __restrict__ nd, float* __restrict__ cat,
                               long long dstBase, int nE) {
    long long tid = (long long)blockIdx.x * blockDim.x + threadIdx.x;
    int e = (int)(tid >> 5);
    if (e >= nE) return;
    int c4 = (int)(tid & 31) * 4;
    int s = src[e], d = dst[e];
    float scale = nd[d];
    if (e + 1 < nE)   // hint the next edge's source row toward the caches
        __builtin_prefetch(&H[(long long)src[e + 1] * 128 + c4], 0, 1);
    float4 hv = *(const float4*)&H[(long long)s * 128 + c4];
    float* p = &cat[dstBase + (long long)d * 128 + c4];
    atomAddF(p + 0, hv.x * scale);
    atomAddF(p + 1, hv.y * scale);
    atomAddF(p + 2, hv.z * scale);
    atomAddF(p + 3, hv.w * scale);
}

// ---------------- final GEMM: out += W_tot @ leaky_relu(cat)  (split-K WMMA) ----------------
// M=256 (4 blocks of 64 rows), N=128 (8 col tiles, one per wave), K split in
// 1024-row chunks over blockIdx.y. leaky_relu applied on B load; partial tiles
// reduced into d_out with f32 atomics (d_out pre-initialized with b_tot).
#define KCHUNK 1024
__global__ __launch_bounds__(256) void final_gemm_kernel(const float* __restrict__ Wt,
                                                         const float* __restrict__ cat,
                                                         float* __restrict__ out) {
    const int t    = threadIdx.x;
    const int lane = t & 31;
    const int wave = t >> 5;
    const int half = lane >> 4;
    const int l16  = lane & 15;
    const int m0   = blockIdx.x * 64;
    const int n0   = wave * 16;
    const int k0   = blockIdx.y * KCHUNK;
    const int kEnd = (k0 + KCHUNK < TOTAL) ? (k0 + KCHUNK) : TOTAL;

    // K-invariant A-row base pointers into W_tot
    const float* Arow[4];
#pragma unroll
    for (int rt = 0; rt < 4; ++rt)
        Arow[rt] = Wt + (long long)(m0 + rt * 16 + l16) * TOTAL;

    v8f acc[4] = {v8f{}, v8f{}, v8f{}, v8f{}};

    for (int k = k0; k < kEnd; k += 4) {
        const int kk = k + 2 * half;
        v2f b;
        b.x = lrelu(cat[(long long)kk * 128 + n0 + l16]);
        b.y = lrelu(cat[(long long)(kk + 1) * 128 + n0 + l16]);
#pragma unroll
        for (int rt = 0; rt < 4; ++rt) {
            v2f a = *(const v2f*)&Arow[rt][kk];
            acc[rt] = __builtin_amdgcn_wmma_f32_16x16x4_f32(
                false, a, false, b, (short)0, acc[rt], false, false);
        }
    }

#pragma unroll
    for (int rt = 0; rt < 4; ++rt) {
#pragma unroll
        for (int i = 0; i < 8; ++i) {
            int m = m0 + rt * 16 + i + 8 * half;
            atomAddF(&out[m * 128 + n0 + l16], acc[rt][i]);
        }
    }
}

// ---------------- launch ----------------
static inline int cdiv(long long a, long long b) { return (int)((a + b - 1) / b); }

extern "C" void kernel_launch(void* const* d_in, const int* in_sizes, int n_in,
                              void* d_out, int out_size, void* d_ws, size_t ws_size,
                              hipStream_t stream) {
    const float* feat_svc  = (const float*)d_in[0];
    const float* feat_inst = (const float*)d_in[1];
    const float* feat_node = (const float*)d_in[2];
    const int*   svc_src   = (const int*)d_in[3];
    const int*   svc_dst   = (const int*)d_in[4];
    const int*   in_src    = (const int*)d_in[5];
    const int*   in_dst    = (const int*)d_in[6];
    const int*   ni_src    = (const int*)d_in[7];
    const int*   ni_dst    = (const int*)d_in[8];
    const int*   ii_src    = (const int*)d_in[9];
    const int*   ii_dst    = (const int*)d_in[10];
    const float* W_sc = (const float*)d_in[11];  const float* b_sc = (const float*)d_in[12];
    const float* W_in = (const float*)d_in[13];  const float* b_in = (const float*)d_in[14];
    const float* W_ni = (const float*)d_in[15];  const float* b_ni = (const float*)d_in[16];
    const float* W_ii = (const float*)d_in[17];  const float* b_ii = (const float*)d_in[18];
    const float* W_tot = (const float*)d_in[19]; const float* b_tot = (const float*)d_in[20];
    float* out = (float*)d_out;

    float* ws   = (float*)d_ws;
    float* cat  = ws + OFF_CAT;
    float* h_sc = ws + OFF_HSC;
    float* h_in = ws + OFF_HIN;
    float* h_ni = ws + OFF_HNI;
    float* h_ii = ws + OFF_HII;
    float* deg  = ws + OFF_DEG;
    float* ns_sc = ws + D_NS_SC; float* nd_sc = ws + D_ND_SC;
    float* ns_in = ws + D_NS_IN; float* nd_in = ws + D_ND_IN;
    float* ns_ni = ws + D_NS_NI; float* nd_ni = ws + D_ND_NI;
    float* ns_ii = ws + D_NS_II; float* nd_ii = ws + D_ND_II;

    // 1) init
    init_deg_kernel<<<cdiv(DEG_TOTAL, 256), 256, 0, stream>>>(deg);
    init_cat_kernel<<<cdiv((long long)TOTAL * 32, 256), 256, 0, stream>>>(cat, b_sc, b_in, b_ni, b_ii);
    init_out_kernel<<<cdiv(OUTD * HDIM, 256), 256, 0, stream>>>(out, b_tot);

    // 2) degrees
    degree_kernel<<<cdiv(E_SC, 256), 256, 0, stream>>>(svc_src, svc_dst, ns_sc, nd_sc, E_SC);
    degree_kernel<<<cdiv(E_IN, 256), 256, 0, stream>>>(in_src,  in_dst,  ns_in, nd_in, E_IN);
    degree_kernel<<<cdiv(E_NI, 256), 256, 0, stream>>>(ni_src,  ni_dst,  ns_ni, nd_ni, E_NI);
    degree_kernel<<<cdiv(E_II, 256), 256, 0, stream>>>(ii_src,  ii_dst,  ns_ii, nd_ii, E_II);
    rsqrt_kernel<<<cdiv(DEG_TOTAL, 256), 256, 0, stream>>>(deg);

    // 3) projections (fp32 WMMA): h = (X @ W) * ns
    proj_kernel<<<cdiv(SVCN, 64),  256, 0, stream>>>(feat_svc,  W_sc, ns_sc, h_sc, SVCN);
    proj_kernel<<<cdiv(INSTN, 64), 256, 0, stream>>>(feat_inst, W_in, ns_in, h_in, INSTN);
    proj_kernel<<<cdiv(NODEN, 64), 256, 0, stream>>>(feat_node, W_ni, ns_ni, h_ni, NODEN);
    proj_kernel<<<cdiv(INSTN, 64), 256, 0, stream>>>(feat_inst, W_ii, ns_ii, h_ii, INSTN);

    // 4) edge scatter with folded destination normalization
    scatter_kernel<<<cdiv((long long)E_SC * 32, 256), 256, 0, stream>>>(
        h_sc, svc_src, svc_dst, nd_sc, cat, 0LL, E_SC);
    scatter_kernel<<<cdiv((long long)E_IN * 32, 256), 256, 0, stream>>>(
        h_in, in_src, in_dst, nd_in, cat, (long long)SVCN * 128, E_IN);
    scatter_kernel<<<cdiv((long long)E_NI * 32, 256), 256, 0, stream>>>(
        h_ni, ni_src, ni_dst, nd_ni, cat, (long long)(SVCN + NODEN) * 128, E_NI);
    scatter_kernel<<<cdiv((long long)E_II * 32, 256), 256, 0, stream>>>(
        h_ii, ii_src, ii_dst, nd_ii, cat, (long long)(SVCN + NODEN) * 128, E_II);

    // 5) final split-K WMMA GEMM with fused leaky_relu
    dim3 g(OUTD / 64, cdiv(TOTAL, KCHUNK));
    final_gemm_kernel<<<g, 256, 0, stream>>>(W_tot, cat, out);
}